// SkeletalUnpool_56959856280300
// MI455X (gfx1250) — compile-verified
//
#include <hip/hip_runtime.h>
#include <hip/hip_bf16.h>
#include <stdint.h>

// out[b, r, t] = x[b, src_idx[r], t]
// x: (B=32, COLS=3584, T=512) f32 ; src_idx: (ROWS=7168) i32 ; out: (32, 7168, 512) f32
// Bandwidth-bound row gather. One wave32 copies one 2048-byte row through LDS
// using the CDNA5 async data mover (ASYNCcnt-tracked), 4x B128 per phase.
// 2D grid (row-group, batch) removes all integer division from the hot path.

#define WAVES_PER_BLOCK 8
#define ROW_BYTES 2048  // 512 floats

__global__ __launch_bounds__(256)
void SkeletalUnpool_gather_async(const float* __restrict__ x,
                                 const int* __restrict__ src_idx,
                                 float* __restrict__ out,
                                 int cols_in,        // 3584 input rows per batch
                                 int rows_out)       // 7168 output rows per batch
{
    __shared__ __align__(16) char smem[WAVES_PER_BLOCK * ROW_BYTES];

    const int tid  = threadIdx.x;
    const int wave = tid >> 5;   // wave32 on gfx1250
    const int lane = tid & 31;

    const int b = blockIdx.y;                        // batch index
    const int r = blockIdx.x * WAVES_PER_BLOCK + wave;  // output row in batch
    if (r >= rows_out) return;   // wave-uniform; EXEC is all-1s inside

    const int src = src_idx[r];  // uniform per wave, L2-resident table

    const unsigned long long gin =
        (unsigned long long)(uintptr_t)(x + ((long long)b * cols_in + src) * 512) +
        (unsigned long long)(lane * 16);
    const unsigned long long gout =
        (unsigned long long)(uintptr_t)(out + ((long long)b * rows_out + r) * 512) +
        (unsigned long long)(lane * 16);

    // Low 32 bits of a generic pointer to LDS == wave-relative LDS byte offset.
    const unsigned lds = (unsigned)(uintptr_t)(&smem[wave * ROW_BYTES + lane * 16]);

    // Async load 2048 B (4 ops x 32 lanes x 16 B). The immediate offset applies
    // to BOTH the global and LDS addresses (CDNA5 ISA).
    asm volatile(
        "global_load_async_to_lds_b128 %0, %1, off\n\t"
        "global_load_async_to_lds_b128 %0, %1, off offset:512\n\t"
        "global_load_async_to_lds_b128 %0, %1, off offset:1024\n\t"
        "global_load_async_to_lds_b128 %0, %1, off offset:1536"
        :
        : "v"(lds), "v"(gin)
        : "memory");

    // Loads must have landed in LDS before the async stores read it.
    asm volatile("s_wait_asynccnt 0x0" ::: "memory");

    asm volatile(
        "global_store_async_from_lds_b128 %0, %1, off\n\t"
        "global_store_async_from_lds_b128 %0, %1, off offset:512\n\t"
        "global_store_async_from_lds_b128 %0, %1, off offset:1024\n\t"
        "global_store_async_from_lds_b128 %0, %1, off offset:1536"
        :
        : "v"(gout), "v"(lds)
        : "memory");

    // s_endpgm performs an implicit wait-idle; nothing else needed.
}

extern "C" void kernel_launch(void* const* d_in, const int* in_sizes, int n_in,
                              void* d_out, int out_size, void* d_ws, size_t ws_size,
                              hipStream_t stream) {
    const float* x       = (const float*)d_in[0];
    const int*   src_idx = (const int*)d_in[1];
    float*       out     = (float*)d_out;

    const int T = 512;
    const int B = 32;
    const int rows_out = in_sizes[1];                 // 7168
    const int cols_in  = in_sizes[0] / (B * T);       // 3584

    const int blocks_x = (rows_out + WAVES_PER_BLOCK - 1) / WAVES_PER_BLOCK;  // 896
    dim3 grid(blocks_x, B, 1);
    SkeletalUnpool_gather_async<<<grid, 256, 0, stream>>>(
        x, src_idx, out, cols_in, rows_out);
}